// PEER_90443421319757
// MI455X (gfx1250) — compile-verified
//
#include <hip/hip_runtime.h>
#include <math.h>

typedef __attribute__((ext_vector_type(2))) float v2f;
typedef __attribute__((ext_vector_type(8))) float v8f;

#define HEADS 8
#define DKEY  256
#define NKEYS 256
#define TOPK  16
#define NB    2
#define NTOK  1024
#define DMODEL 512
#define QO    4096          // 2*HEADS*DKEY
#define MROWS 2048          // NB*NTOK

__device__ __forceinline__ v8f wmma_f32(v2f a, v2f b, v8f c) {
  // V_WMMA_F32_16X16X4_F32 : D(16x16,f32) = A(16x4,f32) x B(4x16,f32) + C
  return __builtin_amdgcn_wmma_f32_16x16x4_f32(false, a, false, b, (short)0, c,
                                               false, false);
}

// ---------------------------------------------------------------------------
// Kernel 1: Q = X(2048x512) @ Wq^T(512x4096). One wave -> 16(M) x 64(N) strip.
// ---------------------------------------------------------------------------
__global__ __launch_bounds__(256) void peer_qproj(const float* __restrict__ X,
                                                  const float* __restrict__ Wq,
                                                  float* __restrict__ Q) {
  int lane = threadIdx.x & 31;
  int wid  = (blockIdx.x * blockDim.x + threadIdx.x) >> 5;  // 0..8191
  int m_t  = wid & 127;   // 128 M-tiles of 16
  int n_s  = wid >> 7;    // 64 N-strips of 64
  int hf = lane >> 4, ln = lane & 15;
  int m0 = m_t << 4, n0 = n_s << 6;

  v8f c0 = {}, c1 = {}, c2 = {}, c3 = {};
  const float* xrow = X  + (size_t)(m0 + ln) * DMODEL + 2 * hf;
  const float* w0 = Wq + (size_t)(n0 +  0 + ln) * DMODEL + 2 * hf;
  const float* w1 = Wq + (size_t)(n0 + 16 + ln) * DMODEL + 2 * hf;
  const float* w2 = Wq + (size_t)(n0 + 32 + ln) * DMODEL + 2 * hf;
  const float* w3 = Wq + (size_t)(n0 + 48 + ln) * DMODEL + 2 * hf;
  for (int k0 = 0; k0 < DMODEL; k0 += 4) {
    v2f a = *(const v2f*)(xrow + k0);
    c0 = wmma_f32(a, *(const v2f*)(w0 + k0), c0);
    c1 = wmma_f32(a, *(const v2f*)(w1 + k0), c1);
    c2 = wmma_f32(a, *(const v2f*)(w2 + k0), c2);
    c3 = wmma_f32(a, *(const v2f*)(w3 + k0), c3);
  }
  float* qout = Q + (size_t)(m0 + 8 * hf) * QO + n0 + ln;
#pragma unroll
  for (int v = 0; v < 8; ++v) {
    qout[(size_t)v * QO +  0] = c0[v];
    qout[(size_t)v * QO + 16] = c1[v];
    qout[(size_t)v * QO + 32] = c2[v];
    qout[(size_t)v * QO + 48] = c3[v];
  }
}

// ---------------------------------------------------------------------------
// Kernel 2 (fused): sim tile (16 tokens x 256 keys) via WMMA into LDS, then
// per-row top-16 (value+index) with wave shuffle arg-max reduction.
// sim[p,s,n,h,k] = sum_d Q[p*N+n, s*2048+h*256+d] * keys[h,k,p,d]
// ---------------------------------------------------------------------------
__global__ __launch_bounds__(128) void peer_sim_topk(const float* __restrict__ Q,
                                                     const float* __restrict__ Keys,
                                                     float* __restrict__ SX,
                                                     int* __restrict__ IX) {
  __shared__ float sim[16 * NKEYS];  // 16 KB
  int lane = threadIdx.x & 31;
  int w    = threadIdx.x >> 5;                // 0..3 -> key-column strip of 64
  int nt = blockIdx.x & 63;
  int h  = (blockIdx.x >> 6) & 7;
  int s  = (blockIdx.x >> 9) & 1;
  int p  = (blockIdx.x >> 10) & 1;
  int hf = lane >> 4, ln = lane & 15;
  int n0 = nt << 4;
  int kc0 = w << 6;

  const float* arow = Q + (size_t)(p * NTOK + n0 + ln) * QO
                        + (s * 2048 + h * DKEY) + 2 * hf;
  const float* kbase = Keys + ((size_t)h * NKEYS * 2 + p) * DKEY + 2 * hf;
  const float* kb0 = kbase + (size_t)(kc0 +  0 + ln) * (2 * DKEY);
  const float* kb1 = kbase + (size_t)(kc0 + 16 + ln) * (2 * DKEY);
  const float* kb2 = kbase + (size_t)(kc0 + 32 + ln) * (2 * DKEY);
  const float* kb3 = kbase + (size_t)(kc0 + 48 + ln) * (2 * DKEY);

  v8f c0 = {}, c1 = {}, c2 = {}, c3 = {};
  for (int d0 = 0; d0 < DKEY; d0 += 4) {
    v2f a = *(const v2f*)(arow + d0);
    c0 = wmma_f32(a, *(const v2f*)(kb0 + d0), c0);
    c1 = wmma_f32(a, *(const v2f*)(kb1 + d0), c1);
    c2 = wmma_f32(a, *(const v2f*)(kb2 + d0), c2);
    c3 = wmma_f32(a, *(const v2f*)(kb3 + d0), c3);
  }
#pragma unroll
  for (int v = 0; v < 8; ++v) {
    sim[(v + 8 * hf) * NKEYS + kc0 +  0 + ln] = c0[v];
    sim[(v + 8 * hf) * NKEYS + kc0 + 16 + ln] = c1[v];
    sim[(v + 8 * hf) * NKEYS + kc0 + 32 + ln] = c2[v];
    sim[(v + 8 * hf) * NKEYS + kc0 + 48 + ln] = c3[v];
  }
  __syncthreads();

  // top-16 of 256 per row; wave w handles rows w*4 .. w*4+3
  for (int rr = 0; rr < 4; ++rr) {
    int r = (w << 2) + rr;
    float vals[8];
#pragma unroll
    for (int i = 0; i < 8; ++i) vals[i] = sim[r * NKEYS + lane + 32 * i];
    float topv = 0.f; int topi = 0;
    for (int t = 0; t < TOPK; ++t) {
      float lv = vals[0]; int li = lane;
#pragma unroll
      for (int i = 1; i < 8; ++i)
        if (vals[i] > lv) { lv = vals[i]; li = lane + 32 * i; }
#pragma unroll
      for (int off = 16; off > 0; off >>= 1) {
        float ov = __shfl_xor(lv, off, 32);
        int   oi = __shfl_xor(li, off, 32);
        if (ov > lv || (ov == lv && oi < li)) { lv = ov; li = oi; }
      }
      if (lane == t) { topv = lv; topi = li; }
      if (lane == (li & 31)) {
        int slot = li >> 5;
#pragma unroll
        for (int i = 0; i < 8; ++i)
          if (i == slot) vals[i] = -__builtin_inff();
      }
    }
    if (lane < TOPK) {
      size_t ob = (((size_t)(p * 2 + s) * NTOK + (n0 + r)) * HEADS + h) * TOPK + lane;
      SX[ob] = topv;
      IX[ob] = topi;
    }
  }
}

// ---------------------------------------------------------------------------
// Kernel 3: product-key combine. One wave per (s,n,h): 256 candidates
// (pos = i*16+j), top-16, softmax. Outputs weight + combined expert index.
// ---------------------------------------------------------------------------
__global__ __launch_bounds__(256) void peer_combine(const float* __restrict__ SX,
                                                    const int* __restrict__ IX,
                                                    float* __restrict__ WGT,
                                                    int* __restrict__ IDX) {
  int lane = threadIdx.x & 31;
  int wid  = (blockIdx.x * blockDim.x + threadIdx.x) >> 5;  // 0..16383
  int h = wid & 7;
  int n = (wid >> 3) & 1023;
  int s = wid >> 13;
  size_t b0 = (((size_t)(0 + s) * NTOK + n) * HEADS + h) * TOPK;  // p=0 block
  size_t b1 = (((size_t)(2 + s) * NTOK + n) * HEADS + h) * TOPK;  // p=1 block

  float sxv = SX[b0 + (lane >> 1)];
  int   ixv = IX[b0 + (lane >> 1)];
  int jb = (lane & 1) * 8;
  float vals[8]; int iyv[8];
#pragma unroll
  for (int i = 0; i < 8; ++i) {
    vals[i] = sxv + SX[b1 + jb + i];
    iyv[i]  = IX[b1 + jb + i];
  }
  float topv = -__builtin_inff(); int topc = 0;
  for (int t = 0; t < TOPK; ++t) {
    float lv = vals[0]; int lpos = lane * 8; int lc = ixv * NKEYS + iyv[0];
#pragma unroll
    for (int i = 1; i < 8; ++i)
      if (vals[i] > lv) { lv = vals[i]; lpos = lane * 8 + i; lc = ixv * NKEYS + iyv[i]; }
#pragma unroll
    for (int off = 16; off > 0; off >>= 1) {
      float ov = __shfl_xor(lv, off, 32);
      int   op = __shfl_xor(lpos, off, 32);
      int   oc = __shfl_xor(lc, off, 32);
      if (ov > lv || (ov == lv && op < lpos)) { lv = ov; lpos = op; lc = oc; }
    }
    if (lane == t) { topv = lv; topc = lc; }
    if (lane == (lpos >> 3)) {
      int slot = lpos & 7;
#pragma unroll
      for (int i = 0; i < 8; ++i)
        if (i == slot) vals[i] = -__builtin_inff();
    }
  }
  // softmax over the 16 selected (lane 0 holds the max by construction)
  float sc = (lane < TOPK) ? topv : -__builtin_inff();
  float mx = __shfl(sc, 0, 32);
  float ex = (lane < TOPK) ? __expf(sc - mx) : 0.f;
  float sum = ex;
#pragma unroll
  for (int off = 16; off > 0; off >>= 1) sum += __shfl_xor(sum, off, 32);
  if (lane < TOPK) {
    size_t ob = (((size_t)s * NTOK + n) * HEADS + h) * TOPK + lane;
    WGT[ob] = ex / sum;
    IDX[ob] = topc;
  }
}

// ---------------------------------------------------------------------------
// Kernel 4: expert gather-dot (bandwidth-critical). Block per token.
// down-proj dot -> exact GELU * softmax weight -> up-proj accumulation.
// ---------------------------------------------------------------------------
__global__ __launch_bounds__(256) void peer_experts(const float* __restrict__ X,
                                                    const float* __restrict__ Wdown,
                                                    const float* __restrict__ Wup,
                                                    const float* __restrict__ WGT,
                                                    const int* __restrict__ IDX,
                                                    float* __restrict__ Out) {
  __shared__ float xs[DMODEL];
  __shared__ float coeff[128];
  __shared__ float wl[128];
  __shared__ int   el[128];
  int tid = threadIdx.x;
  int token = blockIdx.x;  // s*NTOK + n
  const float* xrow = X + (size_t)token * DMODEL;
  xs[tid]       = xrow[tid];
  xs[tid + 256] = xrow[tid + 256];
  if (tid < 128) {
    wl[tid] = WGT[(size_t)token * 128 + tid];
    el[tid] = IDX[(size_t)token * 128 + tid];
  }
  __syncthreads();

  int lane = tid & 31, w = tid >> 5;
  for (int t = 0; t < 16; ++t) {
    int e = w + 8 * t;
    const float* wd = Wdown + (size_t)el[e] * DMODEL;
    float acc = 0.f;
#pragma unroll
    for (int j = 0; j < 16; ++j) acc += xs[lane + 32 * j] * wd[lane + 32 * j];
#pragma unroll
    for (int off = 16; off > 0; off >>= 1) acc += __shfl_xor(acc, off, 32);
    if (lane == 0) {
      float g = 0.5f * acc * (1.f + erff(acc * 0.70710678118654752f));  // exact GELU
      coeff[e] = g * wl[e];
    }
  }
  __syncthreads();

  int d0 = tid * 2;
  float a0 = 0.f, a1 = 0.f;
  for (int e = 0; e < 128; ++e) {
    float c = coeff[e];
    v2f u = *(const v2f*)(Wup + (size_t)el[e] * DMODEL + d0);
    a0 += c * u.x;
    a1 += c * u.y;
  }
  Out[(size_t)token * DMODEL + d0]     = a0;
  Out[(size_t)token * DMODEL + d0 + 1] = a1;
}

// ---------------------------------------------------------------------------
extern "C" void kernel_launch(void* const* d_in, const int* in_sizes, int n_in,
                              void* d_out, int out_size, void* d_ws, size_t ws_size,
                              hipStream_t stream) {
  (void)in_sizes; (void)n_in; (void)out_size; (void)ws_size;
  const float* X    = (const float*)d_in[0];
  const float* Wq   = (const float*)d_in[1];
  const float* Keys = (const float*)d_in[2];
  const float* Wd   = (const float*)d_in[3];
  const float* Wu   = (const float*)d_in[4];
  float* Out = (float*)d_out;

  // workspace layout (floats): Q[2048*4096] | SX[524288] | IX[524288] |
  //                            WGT[262144] | IDX[262144]   (~40 MB total)
  float* Q   = (float*)d_ws;
  float* SX  = Q + (size_t)MROWS * QO;
  int*   IX  = (int*)(SX + (size_t)2 * 2 * NTOK * HEADS * TOPK);
  float* WGT = (float*)(IX + (size_t)2 * 2 * NTOK * HEADS * TOPK);
  int*   IDX = (int*)(WGT + (size_t)NB * NTOK * HEADS * TOPK);

  peer_qproj   <<<1024, 256, 0, stream>>>(X, Wq, Q);
  peer_sim_topk<<<2048, 128, 0, stream>>>(Q, Keys, SX, IX);
  peer_combine <<<2048, 256, 0, stream>>>(SX, IX, WGT, IDX);
  peer_experts <<<2048, 256, 0, stream>>>(X, Wd, Wu, WGT, IDX, Out);
}